// OptiXRoutingWrapper_4638564680455
// MI455X (gfx1250) — compile-verified
//
#include <hip/hip_runtime.h>

typedef __attribute__((ext_vector_type(2))) float v2f;
typedef __attribute__((ext_vector_type(4))) float v4f;
typedef __attribute__((ext_vector_type(8))) float v8f;

#define SHARPNESS 10.0f
#define LOG2E 1.44269504088896340736f
#define SCALE (SHARPNESS * LOG2E)

// One fused DPP16 reduction step: x = op(dpp(x), x). Patterns xor{1,2,7,15}
// are independent over GF(2) and stay inside each 16-lane row, so 4 steps
// reduce across the half-group with zero LDS traffic and zero waits.
#define DPP_STEP(OP, CTRL, X) \
  asm(OP " %0, %0, %0 " CTRL " row_mask:0xf bank_mask:0xf" : "+v"(X))

__device__ __forceinline__ float red_max16(float x) {
  DPP_STEP("v_max_num_f32_dpp", "quad_perm:[1,0,3,2]", x);
  DPP_STEP("v_max_num_f32_dpp", "quad_perm:[2,3,0,1]", x);
  DPP_STEP("v_max_num_f32_dpp", "row_half_mirror", x);
  DPP_STEP("v_max_num_f32_dpp", "row_mirror", x);
  return x;
}
__device__ __forceinline__ float red_min16(float x) {
  DPP_STEP("v_min_num_f32_dpp", "quad_perm:[1,0,3,2]", x);
  DPP_STEP("v_min_num_f32_dpp", "quad_perm:[2,3,0,1]", x);
  DPP_STEP("v_min_num_f32_dpp", "row_half_mirror", x);
  DPP_STEP("v_min_num_f32_dpp", "row_mirror", x);
  return x;
}
__device__ __forceinline__ float red_add16(float x) {
  DPP_STEP("v_add_f32_dpp", "quad_perm:[1,0,3,2]", x);
  DPP_STEP("v_add_f32_dpp", "quad_perm:[2,3,0,1]", x);
  DPP_STEP("v_add_f32_dpp", "row_half_mirror", x);
  DPP_STEP("v_add_f32_dpp", "row_mirror", x);
  return x;
}

// One wave32 = 16 positions x 64 experts via 4x V_WMMA_F32_16X16X4_F32.
// acc = C + A x B = |p|^2 + eps - 2 p.c + |c|^2 = dist^2 directly.
// Logits pre-scaled by log2(e): max/argmax invariant, denominator = sum 2^lgt
// (no max-shift needed: lgt <= 14.43*safe_radius, no overflow; underflow -> 0).
// Forward STE output is 0 except ste=(1+s)-s, s=2^mx/sum, at the argmax.
__global__ __launch_bounds__(256) void optix_route_kernel(
    const float* __restrict__ pos,   // (B,3)
    const float* __restrict__ ctr,   // (64,3)
    const float* __restrict__ rad,   // (64,)
    float* __restrict__ probs,       // (B,64)
    int* __restrict__ ids,           // (B,)
    int B)
{
  const int lane  = threadIdx.x & 31;
  const int wave  = (blockIdx.x << 3) + (threadIdx.x >> 5); // 8 waves/block
  const int baseM = wave << 4;                              // 16 positions/wave
  const int lo    = lane & 15;
  const int half  = lane >> 4;

  // ---- this lane's position (clamped; full EXEC preserved for WMMA) ----
  int pi = baseM + lo;
  if (pi >= B) pi = B - 1;
  const float px = pos[3 * pi + 0];
  const float py = pos[3 * pi + 1];
  const float pz = pos[3 * pi + 2];
  const float n2 = px * px + py * py + pz * pz;

  // A (16x4 f32 ISA layout): lanes 0-15 hold (K0,K1)=(x,y), lanes 16-31 (K2,K3)=(z,1)
  v2f A;
  A.x = half ? pz : px;
  A.y = half ? 1.0f : py;

  // C accumulator rows m = r + 8*half: broadcast |p|^2 from lane (8*half + r)
  const int h32 = half << 5; // byte index of lane 8*half
  v8f c0;
#pragma unroll
  for (int r = 0; r < 8; ++r)
    c0[r] = __int_as_float(
                __builtin_amdgcn_ds_bpermute(h32 + 4 * r, __float_as_int(n2))) +
            1e-12f;

  // float-domain expert indices owned by this lane (exact for 0..63)
  const float flo  = (float)lo;       // index within tile
  const float feb  = 4.0f * flo;      // first of the 4 experts this lane stores

  // ---- 4 expert tiles: WMMA (acc = dist^2) -> log2-scaled logits ----
  float lgt[4][8];
#pragma unroll
  for (int t = 0; t < 4; ++t) {
    const int e = t * 16 + lo;
    const float cx = ctr[3 * e + 0];
    const float cy = ctr[3 * e + 1];
    const float cz = ctr[3 * e + 2];
    const float cn2 = cx * cx + cy * cy + cz * cz;
    const float rK  = SCALE * fmaxf(fabsf(rad[e]), 0.01f); // log2e*10*safe_radius

    v2f Bm;
    Bm.x = half ? (-2.0f * cz) : (-2.0f * cx);
    Bm.y = half ? cn2 : (-2.0f * cy);

    v8f acc = __builtin_amdgcn_wmma_f32_16x16x4_f32(
        false, A, false, Bm, (short)0, c0, false, false);

#pragma unroll
    for (int r = 0; r < 8; ++r) {
      const float dist = __builtin_amdgcn_sqrtf(acc[r]);  // raw v_sqrt_f32
      lgt[t][r] = fmaf(-SCALE, dist, rK);                 // log2e*10*(rs - dist)
    }
  }

  // ---- per-row max / first-argmax / 2^x-sum via DPP16 butterflies ----
#pragma unroll
  for (int r = 0; r < 8; ++r) {
    // denominator (independent of max -> overlaps the max butterfly)
    float sum = __builtin_amdgcn_exp2f(lgt[0][r]) + __builtin_amdgcn_exp2f(lgt[1][r]) +
                __builtin_amdgcn_exp2f(lgt[2][r]) + __builtin_amdgcn_exp2f(lgt[3][r]);
    sum = red_add16(sum);

    float mx = fmaxf(fmaxf(lgt[0][r], lgt[1][r]), fmaxf(lgt[2][r], lgt[3][r]));
    mx = red_max16(mx);

    // first-argmax = min float index where logit == mx (descending t -> lowest e in-lane)
    float fidx = 1e30f;
#pragma unroll
    for (int t = 3; t >= 0; --t)
      fidx = (lgt[t][r] == mx) ? (flo + (float)(16 * t)) : fidx;
    fidx = red_min16(fidx);

    const float s   = __builtin_amdgcn_exp2f(mx) * __builtin_amdgcn_rcpf(sum);
    const float ste = (1.0f + s) - s;              // STE forward value at argmax
    const int   idx = (int)fidx;

    const int m = baseM + r + 8 * half;
    if (m < B) {
      // lane lo owns experts [4*lo, 4*lo+3] of row m -> one NT b128 store
      v4f v;
      v.x = (fidx == feb + 0.0f) ? ste : 0.0f;
      v.y = (fidx == feb + 1.0f) ? ste : 0.0f;
      v.z = (fidx == feb + 2.0f) ? ste : 0.0f;
      v.w = (fidx == feb + 3.0f) ? ste : 0.0f;
      __builtin_nontemporal_store(v, (v4f*)(probs + (size_t)m * 64 + (lo << 2)));
      if (lo == r) __builtin_nontemporal_store(idx, ids + m);
    }
  }
}

extern "C" void kernel_launch(void* const* d_in, const int* in_sizes, int n_in,
                              void* d_out, int out_size, void* d_ws, size_t ws_size,
                              hipStream_t stream) {
  const float* pos = (const float*)d_in[0]; // positions_3d (B,3)
  const float* ctr = (const float*)d_in[1]; // centers (64,3)
  const float* rad = (const float*)d_in[2]; // radii (64,)
  const int B = in_sizes[0] / 3;

  float* probs = (float*)d_out;                    // (B,64) float32
  int*   ids   = (int*)(probs + (size_t)B * 64);   // (B,) int32, concatenated

  const int blocks = (B + 127) / 128;              // 128 positions per 256-thread block
  optix_route_kernel<<<blocks, 256, 0, stream>>>(pos, ctr, rad, probs, ids, B);
}